// RydbergEvolver_63264868270574
// MI455X (gfx1250) — compile-verified
//
#include <hip/hip_runtime.h>
#include <math.h>

typedef float v2f __attribute__((ext_vector_type(2)));
typedef float v8f __attribute__((ext_vector_type(8)));

#define NQ 9
#define DD 512
#define BATCH 64
#define HID 130
#define DIMX 256
#define PLANE (DD * DD)            // 262144 floats per plane
#define IW (866.0f / 729.0f)       // 866 / 3^6
#define TWO_PI 6.283185307179586f
#define TAYLOR_K 12
#define SMAX 12

__device__ __forceinline__ float softplus_f(float x) {
    return (x > 30.0f) ? x : log1pf(expf(x));
}

// ---------------------------------------------------------------- MLP ------
__global__ __launch_bounds__(256) void mlp_kernel(
    const float* __restrict__ x, const float* __restrict__ W1,
    const float* __restrict__ b1, const float* __restrict__ W2,
    const float* __restrict__ b2, float* __restrict__ params) {
    int b = blockIdx.x, tid = threadIdx.x;
    __shared__ float xs[DIMX];
    __shared__ float hid[HID];
    __shared__ float vv[4];
    xs[tid] = x[b * DIMX + tid];
    __syncthreads();
    if (tid < HID) {
        float acc = b1[tid];
        for (int d = 0; d < DIMX; ++d) acc += xs[d] * W1[d * HID + tid];
        hid[tid] = fmaxf(acc, 0.0f);
    }
    __syncthreads();
    if (tid < 4) {
        float acc = b2[tid];
        for (int h = 0; h < HID; ++h) acc += hid[h] * W2[h * 4 + tid];
        vv[tid] = acc;
    }
    __syncthreads();
    if (tid == 0) {
        float omega = softplus_f(vv[0]);
        float delta = vv[1];
        float phi   = vv[2] - TWO_PI * floorf(vv[2] / TWO_PI);
        float t     = softplus_f(vv[3]);
        float* p = params + b * 8;
        p[0] = omega; p[1] = delta; p[2] = phi; p[3] = t;
    }
}

// ------------------------------------------------------- diagonal + norm ---
__global__ __launch_bounds__(256) void diag_kernel(
    const float* __restrict__ mask, const float* __restrict__ adj,
    float* __restrict__ params, float* __restrict__ diagbuf,
    int* __restrict__ sb) {
    int b = blockIdx.x, tid = threadIdx.x;
    __shared__ float m[NQ];
    __shared__ float red[256];
    if (tid < NQ) m[tid] = mask[b * NQ + tid];
    __syncthreads();
    float omega = params[b * 8 + 0];
    float delta = params[b * 8 + 1];
    float phi   = params[b * 8 + 2];
    float t     = params[b * 8 + 3];
    float lmax = 0.0f;
    for (int s = tid; s < DD; s += 256) {
        float u[NQ]; float sumo = 0.0f;
        for (int i = 0; i < NQ; ++i) {
            float occ = 1.0f - (float)((s >> (NQ - 1 - i)) & 1);
            u[i] = m[i] * occ; sumo += u[i];
        }
        float inter = 0.0f;
        for (int i = 0; i < NQ; ++i) {
            float acc = 0.0f;
            for (int j = 0; j < NQ; ++j) acc += adj[i * NQ + j] * u[j];
            inter += u[i] * acc;
        }
        float d = IW * inter - delta * sumo;
        diagbuf[b * DD + s] = d;
        lmax = fmaxf(lmax, fabsf(d));
    }
    red[tid] = lmax;
    __syncthreads();
    for (int o = 128; o > 0; o >>= 1) {
        if (tid < o) red[tid] = fmaxf(red[tid], red[tid + o]);
        __syncthreads();
    }
    if (tid == 0) {
        float summ = 0.0f;
        for (int j = 0; j < NQ; ++j) summ += m[j];
        float norm1 = red[0] + 0.5f * omega * summ;
        float r = t * norm1;
        int sv = 0;
        if (r > 1.0f) {
            sv = (int)ceilf(log2f(r));
            sv = sv < 0 ? 0 : (sv > SMAX ? SMAX : sv);
        }
        float tau = t * exp2f(-(float)sv);
        params[b * 8 + 4] = tau;
        params[b * 8 + 5] = 0.5f * omega * cosf(phi);   // hc
        params[b * 8 + 6] = 0.5f * omega * sinf(phi);   // hs
        sb[b] = sv;
    }
}

// ------------------------------------------------- build A = -i*tau*H ------
__global__ __launch_bounds__(256) void build_kernel(
    const float* __restrict__ mask, const float* __restrict__ params,
    const float* __restrict__ diagbuf, float* __restrict__ Abase) {
    int b = blockIdx.y;
    int idx = blockIdx.x * 256 + threadIdx.x;     // 0 .. 262143
    int p = idx >> 9, q = idx & (DD - 1);
    float tau = params[b * 8 + 4];
    float hc  = params[b * 8 + 5];
    float hs  = params[b * 8 + 6];
    float Hr = 0.0f, Hi = 0.0f;
    int x = p ^ q;
    if (x == 0) {
        Hr = diagbuf[b * DD + p];
    } else if ((x & (x - 1)) == 0) {
        int beta = __ffs(x) - 1;          // bit position
        int j = NQ - 1 - beta;            // qubit index
        float mj = mask[b * NQ + j];
        Hr = mj * hc;
        Hi = ((p >> beta) & 1) ? -mj * hs : mj * hs;
    }
    float* Ar = Abase + (size_t)b * 2 * PLANE;
    float* Ai = Ar + PLANE;
    // (-i)(Hr + i Hi) = Hi - i Hr
    Ar[idx] =  tau * Hi;
    Ai[idx] = -tau * Hr;
}

// ----------------------------------------- Y = I + A / TAYLOR_K  (init) ----
__global__ __launch_bounds__(256) void init_kernel(
    const float* __restrict__ Abase, float* __restrict__ Ybase) {
    int b = blockIdx.y;
    int idx = blockIdx.x * 256 + threadIdx.x;
    int p = idx >> 9, q = idx & (DD - 1);
    const float* Ar = Abase + (size_t)b * 2 * PLANE;
    float* Yr = Ybase + (size_t)b * 2 * PLANE;
    float invK = 1.0f / (float)TAYLOR_K;
    Yr[idx]         = Ar[idx] * invK + ((p == q) ? 1.0f : 0.0f);
    Yr[idx + PLANE] = Ar[idx + PLANE] * invK;
}

// -------------------------------------------- complex GEMM via f32 WMMA ----
// 2x2 register-blocked: each wave computes a 32x32 output region.
// mode 0: C = alpha * (A @ B) + add_id * I       (Horner step)
// mode 1: C = (it < sb[b]) ? B @ B : B           (conditional squaring)
__global__ __launch_bounds__(256) void zgemm_kernel(
    const float* __restrict__ Abase, const float* __restrict__ Bbase,
    float* __restrict__ Cbase, float alpha, int add_id, int mode, int it,
    const int* __restrict__ sb) {
    int b = blockIdx.y;
    const float* Ar = Abase + (size_t)b * 2 * PLANE;
    const float* Ai = Ar + PLANE;
    const float* Br = Bbase + (size_t)b * 2 * PLANE;
    const float* Bi = Br + PLANE;
    float* Cr = Cbase + (size_t)b * 2 * PLANE;
    float* Ci = Cr + PLANE;

    int lane = threadIdx.x;                  // 0..31 (wave32)
    int wave = threadIdx.y;                  // 0..7
    int region = blockIdx.x * 8 + wave;      // 0..255 (16x16 regions of 32x32)
    int rm = (region >> 4) * 32;
    int rn = (region & 15) * 32;
    int lhalf = lane >> 4;                   // 0 or 1
    int l15 = lane & 15;

    if (mode == 1 && it >= sb[b]) {          // inactive batch: copy B -> C
        for (int mi = 0; mi < 2; ++mi)
            for (int ni = 0; ni < 2; ++ni)
                for (int r = 0; r < 8; ++r) {
                    int mg = rm + mi * 16 + r + lhalf * 8;
                    int nc = rn + ni * 16 + l15;
                    Cr[(size_t)mg * DD + nc] = Br[(size_t)mg * DD + nc];
                    Ci[(size_t)mg * DD + nc] = Bi[(size_t)mg * DD + nc];
                }
        return;
    }

    // A fragment (16x4 f32): lanes 0-15 -> K = k0,k0+1 ; lanes 16-31 -> +2,+3
    int arow0 = rm + l15;
    int arow1 = rm + 16 + l15;
    int nc0 = rn + l15;
    int nc1 = rn + 16 + l15;
    int kofs = lhalf * 2;

    v8f pr00 = {}, pr01 = {}, pr10 = {}, pr11 = {};
    v8f pi00 = {}, pi01 = {}, pi10 = {}, pi11 = {};

    for (int k0 = 0; k0 < DD; k0 += 4) {
        int ka = k0 + kofs;
        v2f ar0 = *(const v2f*)(Ar + (size_t)arow0 * DD + ka);
        v2f ai0 = *(const v2f*)(Ai + (size_t)arow0 * DD + ka);
        v2f ar1 = *(const v2f*)(Ar + (size_t)arow1 * DD + ka);
        v2f ai1 = *(const v2f*)(Ai + (size_t)arow1 * DD + ka);
        v2f br0, bi0, br1, bi1;
        br0.x = Br[(size_t)ka * DD + nc0];
        br0.y = Br[(size_t)(ka + 1) * DD + nc0];
        bi0.x = Bi[(size_t)ka * DD + nc0];
        bi0.y = Bi[(size_t)(ka + 1) * DD + nc0];
        br1.x = Br[(size_t)ka * DD + nc1];
        br1.y = Br[(size_t)(ka + 1) * DD + nc1];
        bi1.x = Bi[(size_t)ka * DD + nc1];
        bi1.y = Bi[(size_t)(ka + 1) * DD + nc1];
        v2f an0; an0.x = -ai0.x; an0.y = -ai0.y;
        v2f an1; an1.x = -ai1.x; an1.y = -ai1.y;

        pr00 = __builtin_amdgcn_wmma_f32_16x16x4_f32(false, ar0, false, br0,
                                                     (short)0, pr00, false, false);
        pr00 = __builtin_amdgcn_wmma_f32_16x16x4_f32(false, an0, false, bi0,
                                                     (short)0, pr00, false, false);
        pi00 = __builtin_amdgcn_wmma_f32_16x16x4_f32(false, ar0, false, bi0,
                                                     (short)0, pi00, false, false);
        pi00 = __builtin_amdgcn_wmma_f32_16x16x4_f32(false, ai0, false, br0,
                                                     (short)0, pi00, false, false);

        pr01 = __builtin_amdgcn_wmma_f32_16x16x4_f32(false, ar0, false, br1,
                                                     (short)0, pr01, false, false);
        pr01 = __builtin_amdgcn_wmma_f32_16x16x4_f32(false, an0, false, bi1,
                                                     (short)0, pr01, false, false);
        pi01 = __builtin_amdgcn_wmma_f32_16x16x4_f32(false, ar0, false, bi1,
                                                     (short)0, pi01, false, false);
        pi01 = __builtin_amdgcn_wmma_f32_16x16x4_f32(false, ai0, false, br1,
                                                     (short)0, pi01, false, false);

        pr10 = __builtin_amdgcn_wmma_f32_16x16x4_f32(false, ar1, false, br0,
                                                     (short)0, pr10, false, false);
        pr10 = __builtin_amdgcn_wmma_f32_16x16x4_f32(false, an1, false, bi0,
                                                     (short)0, pr10, false, false);
        pi10 = __builtin_amdgcn_wmma_f32_16x16x4_f32(false, ar1, false, bi0,
                                                     (short)0, pi10, false, false);
        pi10 = __builtin_amdgcn_wmma_f32_16x16x4_f32(false, ai1, false, br0,
                                                     (short)0, pi10, false, false);

        pr11 = __builtin_amdgcn_wmma_f32_16x16x4_f32(false, ar1, false, br1,
                                                     (short)0, pr11, false, false);
        pr11 = __builtin_amdgcn_wmma_f32_16x16x4_f32(false, an1, false, bi1,
                                                     (short)0, pr11, false, false);
        pi11 = __builtin_amdgcn_wmma_f32_16x16x4_f32(false, ar1, false, bi1,
                                                     (short)0, pi11, false, false);
        pi11 = __builtin_amdgcn_wmma_f32_16x16x4_f32(false, ai1, false, br1,
                                                     (short)0, pi11, false, false);
    }

    for (int r = 0; r < 8; ++r) {
        int mg0 = rm + r + lhalf * 8;
        int mg1 = rm + 16 + r + lhalf * 8;
        float v00r = pr00[r] * alpha, v00i = pi00[r] * alpha;
        float v01r = pr01[r] * alpha, v01i = pi01[r] * alpha;
        float v10r = pr10[r] * alpha, v10i = pi10[r] * alpha;
        float v11r = pr11[r] * alpha, v11i = pi11[r] * alpha;
        if (add_id) {
            if (mg0 == nc0) v00r += 1.0f;
            if (mg0 == nc1) v01r += 1.0f;
            if (mg1 == nc0) v10r += 1.0f;
            if (mg1 == nc1) v11r += 1.0f;
        }
        Cr[(size_t)mg0 * DD + nc0] = v00r; Ci[(size_t)mg0 * DD + nc0] = v00i;
        Cr[(size_t)mg0 * DD + nc1] = v01r; Ci[(size_t)mg0 * DD + nc1] = v01i;
        Cr[(size_t)mg1 * DD + nc0] = v10r; Ci[(size_t)mg1 * DD + nc0] = v10i;
        Cr[(size_t)mg1 * DD + nc1] = v11r; Ci[(size_t)mg1 * DD + nc1] = v11i;
    }
}

// ------------------------------------------------ extract column 0 ---------
__global__ __launch_bounds__(256) void extract_kernel(
    const float* __restrict__ Ybase, float* __restrict__ out) {
    int idx = blockIdx.x * 256 + threadIdx.x;   // 0 .. 32767  (b*512 + p)
    int b = idx >> 9, p = idx & (DD - 1);
    const float* Yr = Ybase + (size_t)b * 2 * PLANE;
    out[idx * 2]     = Yr[(size_t)p * DD];
    out[idx * 2 + 1] = Yr[PLANE + (size_t)p * DD];
}

// ---------------------------------------------------------------------------
extern "C" void kernel_launch(void* const* d_in, const int* in_sizes, int n_in,
                              void* d_out, int out_size, void* d_ws,
                              size_t ws_size, hipStream_t stream) {
    const float* x    = (const float*)d_in[0];
    const float* mask = (const float*)d_in[1];
    const float* adj  = (const float*)d_in[2];
    const float* W1   = (const float*)d_in[3];
    const float* b1   = (const float*)d_in[4];
    const float* W2   = (const float*)d_in[5];
    const float* b2   = (const float*)d_in[6];
    float* out = (float*)d_out;

    float* wsf    = (float*)d_ws;
    float* params = wsf;                         // 64*8
    int*   sb     = (int*)(wsf + 512);           // 64
    float* diagb  = wsf + 1024;                  // 64*512
    float* A      = wsf + 65536;                 // 64 * 2 * PLANE
    float* Y0     = A  + (size_t)BATCH * 2 * PLANE;
    float* Y1     = Y0 + (size_t)BATCH * 2 * PLANE;

    mlp_kernel<<<BATCH, 256, 0, stream>>>(x, W1, b1, W2, b2, params);
    diag_kernel<<<BATCH, 256, 0, stream>>>(mask, adj, params, diagb, sb);
    build_kernel<<<dim3(PLANE / 256, BATCH), 256, 0, stream>>>(mask, params,
                                                               diagb, A);
    init_kernel<<<dim3(PLANE / 256, BATCH), 256, 0, stream>>>(A, Y0);

    dim3 ggrid(32, BATCH);        // 256 regions / 8 waves per block
    dim3 gblk(32, 8);
    float* cur = Y0; float* oth = Y1;
    for (int k = TAYLOR_K - 1; k >= 1; --k) {    // Horner: Y = I + (A Y)/k
        zgemm_kernel<<<ggrid, gblk, 0, stream>>>(A, cur, oth, 1.0f / (float)k,
                                                 1, 0, 0, sb);
        float* t_ = cur; cur = oth; oth = t_;
    }
    for (int it = 0; it < SMAX; ++it) {          // conditional squaring
        zgemm_kernel<<<ggrid, gblk, 0, stream>>>(cur, cur, oth, 1.0f,
                                                 0, 1, it, sb);
        float* t_ = cur; cur = oth; oth = t_;
    }
    extract_kernel<<<BATCH * DD / 256, 256, 0, stream>>>(cur, out);
}